// DifferentiableVCPCBFQP_90580860272651
// MI455X (gfx1250) — compile-verified
//
#include <hip/hip_runtime.h>
#include <math.h>

#define BLOCK   64
#define NC      14
#define NOBS    5
#define NITERS  30

// Per-element CBF-QP interior point solve. One thread = one batch element.
// CDNA5 paths used: async global->LDS staging (ASYNCcnt, b128), wave32 blocks.
// WMMA is not applicable: 32K independent 14x2 QPs solved in f64 (no shared
// matmul structure; CDNA5 WMMA has no f64 path).
__global__ __launch_bounds__(BLOCK) void cbf_qp_kernel(
    const float* __restrict__ u_nom,     // [B,2]
    const float* __restrict__ states,    // [B,3]
    const float* __restrict__ obs_xy,    // [5,2]
    const float* __restrict__ obs_r,     // [5]
    const float* __restrict__ opp,       // [B,3]
    float* __restrict__ out,             // [B,2]
    int B)
{
    // LDS layout (floats): [0, 2B) u_nominal | [2B, 5B) states | [5B, 8B) opponent
    __shared__ __align__(16) float smem[8 * BLOCK];

    const int t   = threadIdx.x;
    const int gid = blockIdx.x * BLOCK + t;

    // ---------------- CDNA5 async global -> LDS staging (b128) ----------------
    {
        // u_nominal slice: BLOCK*2 floats = 512B = 32 x b128
        if (t < (BLOCK * 8) / 16) {
            unsigned go   = (unsigned)(blockIdx.x * (BLOCK * 8)) + (unsigned)(t * 16);
            unsigned gmax = (unsigned)(B * 8 - 16);
            if (go > gmax) go = gmax;
            int lds = t * 16;
            asm volatile("global_load_async_to_lds_b128 %0, %1, %2"
                         :: "v"(lds), "v"(go), "s"(u_nom) : "memory");
        }
        // states / opponent slices: BLOCK*3 floats = 768B = 48 x b128 each
        if (t < (BLOCK * 12) / 16) {
            unsigned go   = (unsigned)(blockIdx.x * (BLOCK * 12)) + (unsigned)(t * 16);
            unsigned gmax = (unsigned)(B * 12 - 16);
            if (go > gmax) go = gmax;
            int lds_s = (2 * BLOCK) * 4 + t * 16;
            int lds_o = (5 * BLOCK) * 4 + t * 16;
            asm volatile("global_load_async_to_lds_b128 %0, %1, %2"
                         :: "v"(lds_s), "v"(go), "s"(states) : "memory");
            asm volatile("global_load_async_to_lds_b128 %0, %1, %2"
                         :: "v"(lds_o), "v"(go), "s"(opp) : "memory");
        }
        asm volatile("s_wait_asynccnt 0" ::: "memory");
    }
    __syncthreads();
    if (gid >= B) return;

    // ---------------- read staged inputs from LDS ----------------
    const float un0 = smem[2 * t + 0];
    const float un1 = smem[2 * t + 1];
    const float sx  = smem[2 * BLOCK + 3 * t + 0];
    const float sy  = smem[2 * BLOCK + 3 * t + 1];
    const float sth = smem[2 * BLOCK + 3 * t + 2];
    const float ox_ = smem[5 * BLOCK + 3 * t + 0];
    const float oy_ = smem[5 * BLOCK + 3 * t + 1];
    const float oth = smem[5 * BLOCK + 3 * t + 2];

    // ---------------- build G, h in f32 (matches reference) ----------------
    const float ct = cosf(sth), st = sinf(sth);
    const float qx = sx + 0.1f * ct;
    const float qy = sy + 0.1f * st;
    const float xl = 10.0f - 0.15f;
    const float yl = 10.0f - 0.15f;

    float g0f[NC], g1f[NC], hf[NC];
    g0f[0] =  ct; g1f[0] = -0.1f * st; hf[0] = xl - qx;
    g0f[1] = -ct; g1f[1] =  0.1f * st; hf[1] = qx + xl;
    g0f[2] =  st; g1f[2] =  0.1f * ct; hf[2] = yl - qy;
    g0f[3] = -st; g1f[3] = -0.1f * ct; hf[3] = qy + yl;

    #pragma unroll
    for (int k = 0; k < NOBS; ++k) {
        float obx = obs_xy[2 * k], oby = obs_xy[2 * k + 1]; // uniform -> SMEM
        float er  = obs_r[k] + 0.15f;
        float dx = qx - obx, dy = qy - oby;
        hf[4 + k] = dx * dx + dy * dy - er * er;
        float a_v  = 2.0f * (dx * ct + dy * st);
        float a_om = 2.0f * (-dx * 0.1f * st + dy * 0.1f * ct);
        g0f[4 + k] = -a_v; g1f[4 + k] = -a_om;
    }
    {
        float cto = cosf(oth), sto = sinf(oth);
        float qxo = ox_ + 0.1f * cto, qyo = oy_ + 0.1f * sto;
        float dxc = qx - qxo, dyc = qy - qyo;
        hf[9] = dxc * dxc + dyc * dyc - 0.35f * 0.35f;
        float a_v  = 2.0f * (dxc * ct + dyc * st);
        float a_om = 2.0f * (-dxc * 0.1f * st + dyc * 0.1f * ct);
        g0f[9] = -a_v; g1f[9] = -a_om;
    }
    g0f[10] = -1.f; g1f[10] =  0.f; hf[10] = -0.0f;  // -V_MIN
    g0f[11] =  1.f; g1f[11] =  0.f; hf[11] =  1.0f;  //  V_MAX
    g0f[12] =  0.f; g1f[12] = -1.f; hf[12] =  2.84f; // -W_MIN
    g0f[13] =  0.f; g1f[13] =  1.f; hf[13] =  2.84f; //  W_MAX

    // ---------------- f64 interior point solve (matches reference) ----------------
    double g0[NC], g1[NC], h[NC], s[NC], lam[NC];
    #pragma unroll
    for (int m = 0; m < NC; ++m) {
        g0[m] = (double)g0f[m]; g1[m] = (double)g1f[m]; h[m] = (double)hf[m];
    }

    const double Q00 = 300.0, Q11 = 2.0;
    const double p0 = -300.0 * (double)un0;
    const double p1 = -2.0   * (double)un1;
    double u0 = (double)un0, u1 = (double)un1;   // Q^{-1}(-p) == u_nominal

    #pragma unroll
    for (int m = 0; m < NC; ++m) {
        double sv = h[m] - (g0[m] * u0 + g1[m] * u1);
        s[m]   = sv > 1.0 ? sv : 1.0;
        lam[m] = 1.0;
    }

    for (int it = 0; it < NITERS; ++it) {
        double inv_s[NC], rp[NC];
        double rd0 = Q00 * u0 + p0;
        double rd1 = Q11 * u1 + p1;
        double M00 = Q00, M01 = 0.0, M11 = Q11;
        double mu = 0.0, t0 = 0.0, t1 = 0.0;

        #pragma unroll
        for (int m = 0; m < NC; ++m) {
            inv_s[m] = 1.0 / s[m];
            double Dg = lam[m] * inv_s[m];
            rd0 += g0[m] * lam[m];
            rd1 += g1[m] * lam[m];
            mu  += s[m] * lam[m];
            rp[m] = g0[m] * u0 + g1[m] * u1 + s[m] - h[m];
            M00 += Dg * g0[m] * g0[m];
            M01 += Dg * g0[m] * g1[m];
            M11 += Dg * g1[m] * g1[m];
            double w = Dg * rp[m] - lam[m];   // Dg*r_p - rc/s with rc = s*lam
            t0 += g0[m] * w; t1 += g1[m] * w;
        }
        mu *= (1.0 / NC);
        const double invdet = 1.0 / (M00 * M11 - M01 * M01);
        double rhs0 = -rd0 - t0, rhs1 = -rd1 - t1;
        const double dua0 = (M11 * rhs0 - M01 * rhs1) * invdet;
        const double dua1 = (M00 * rhs1 - M01 * rhs0) * invdet;

        // affine step length: track min ratio as a fraction (bn/bd), seed = 1.0
        double bn = 1.0, bd = 1.0;
        #pragma unroll
        for (int m = 0; m < NC; ++m) {
            double Dg  = lam[m] * inv_s[m];
            double gdu = g0[m] * dua0 + g1[m] * dua1;
            double dsa = -rp[m] - gdu;
            double dla = -lam[m] + Dg * (rp[m] + gdu);
            if (dsa < 0.0) {               // candidate s/(-dsa), num>0, den>0
                double n = s[m], d = -dsa;
                if (n * bd < bn * d) { bn = n; bd = d; }
            }
            if (dla < 0.0) {               // candidate lam/(-dla)
                double n = lam[m], d = -dla;
                if (n * bd < bn * d) { bn = n; bd = d; }
            }
        }
        const double a_aff = bn / bd;       // single divide

        double mu_aff = 0.0;
        #pragma unroll
        for (int m = 0; m < NC; ++m) {
            double Dg  = lam[m] * inv_s[m];
            double gdu = g0[m] * dua0 + g1[m] * dua1;
            double dsa = -rp[m] - gdu;
            double dla = -lam[m] + Dg * (rp[m] + gdu);
            mu_aff += (s[m] + a_aff * dsa) * (lam[m] + a_aff * dla);
        }
        mu_aff *= (1.0 / NC);
        double ratio = mu_aff / mu;
        const double smu = ratio * ratio * ratio * mu;  // sigma * mu

        // corrector rhs
        t0 = 0.0; t1 = 0.0;
        #pragma unroll
        for (int m = 0; m < NC; ++m) {
            double Dg  = lam[m] * inv_s[m];
            double gdu = g0[m] * dua0 + g1[m] * dua1;
            double dsa = -rp[m] - gdu;
            double dla = -lam[m] + Dg * (rp[m] + gdu);
            double rc  = s[m] * lam[m] + dsa * dla - smu;
            double w   = Dg * rp[m] - rc * inv_s[m];
            t0 += g0[m] * w; t1 += g1[m] * w;
        }
        rhs0 = -rd0 - t0; rhs1 = -rd1 - t1;
        const double du0 = (M11 * rhs0 - M01 * rhs1) * invdet;
        const double du1 = (M00 * rhs1 - M01 * rhs0) * invdet;

        // final step length (fraction-min again)
        bn = 1.0; bd = 1.0;
        #pragma unroll
        for (int m = 0; m < NC; ++m) {
            double Dg    = lam[m] * inv_s[m];
            double gdu_a = g0[m] * dua0 + g1[m] * dua1;
            double dsa   = -rp[m] - gdu_a;
            double dla   = -lam[m] + Dg * (rp[m] + gdu_a);
            double rc    = s[m] * lam[m] + dsa * dla - smu;
            double gdu   = g0[m] * du0 + g1[m] * du1;
            double dsn   = -rp[m] - gdu;
            double dln   = -rc * inv_s[m] + Dg * (rp[m] + gdu);
            if (dsn < 0.0) {
                double n = s[m], d = -dsn;
                if (n * bd < bn * d) { bn = n; bd = d; }
            }
            if (dln < 0.0) {
                double n = lam[m], d = -dln;
                if (n * bd < bn * d) { bn = n; bd = d; }
            }
        }
        const double a = 0.99 * (bn / bd);

        u0 += a * du0; u1 += a * du1;
        #pragma unroll
        for (int m = 0; m < NC; ++m) {
            double Dg    = lam[m] * inv_s[m];
            double gdu_a = g0[m] * dua0 + g1[m] * dua1;
            double dsa   = -rp[m] - gdu_a;
            double dla   = -lam[m] + Dg * (rp[m] + gdu_a);
            double rc    = s[m] * lam[m] + dsa * dla - smu;
            double gdu   = g0[m] * du0 + g1[m] * du1;
            double dsn   = -rp[m] - gdu;
            double dln   = -rc * inv_s[m] + Dg * (rp[m] + gdu);
            s[m]   += a * dsn;
            lam[m] += a * dln;
        }
    }

    // ---------------- clip + coalesced f32 store ----------------
    float r0 = (float)u0; r0 = fminf(fmaxf(r0, 0.0f),   1.0f);
    float r1 = (float)u1; r1 = fminf(fmaxf(r1, -2.84f), 2.84f);
    reinterpret_cast<float2*>(out)[gid] = make_float2(r0, r1);
}

extern "C" void kernel_launch(void* const* d_in, const int* in_sizes, int n_in,
                              void* d_out, int out_size, void* d_ws, size_t ws_size,
                              hipStream_t stream) {
    const float* u_nom  = (const float*)d_in[0];
    const float* states = (const float*)d_in[1];
    const float* oxy    = (const float*)d_in[2];
    const float* orr    = (const float*)d_in[3];
    const float* opp    = (const float*)d_in[4];
    float* out = (float*)d_out;

    const int B    = in_sizes[0] / 2;            // 32768
    const int grid = (B + BLOCK - 1) / BLOCK;    // 512 workgroups of 2 wave32s
    cbf_qp_kernel<<<grid, BLOCK, 0, stream>>>(u_nom, states, oxy, orr, opp, out, B);
}